// GCNLayer_sp_73924977098826
// MI455X (gfx1250) — compile-verified
//
#include <hip/hip_runtime.h>

// GCN sparse aggregation for MI455X (gfx1250, wave32).
// out[rows[e]] += vals[e] * embeds[cols[e]]   (E=1.6M, D=64, fp32)
//
// Memory-bound op (0.25 FLOP/byte): WMMA has no dense structure to exploit.
// CDNA5 paths used instead:
//   - GLOBAL_LOAD_ASYNC_TO_LDS_B128 (ASYNCcnt) to stage edge tiles in LDS
//   - s_wait_asynccnt 0 (gfx1250 split counter)
//   - coalesced 256B global_load_b128 gathers (16 lanes x float4 per row)
//   - native no-return global_atomic_add_f32 scatter into L2-resident output

typedef int v4i __attribute__((ext_vector_type(4)));
typedef __attribute__((address_space(3))) v4i lv4i;   // LDS-qualified int4

constexpr int TILE    = 1024;  // edges per block tile
constexpr int THREADS = 256;   // 8 waves (wave32)

__global__ __launch_bounds__(256)
void gcn_zero_kernel(float4* __restrict__ out4, int n4) {
  int i = blockIdx.x * blockDim.x + threadIdx.x;
  if (i < n4) out4[i] = make_float4(0.f, 0.f, 0.f, 0.f);
}

__global__ __launch_bounds__(THREADS)
void gcn_agg_kernel(const int* __restrict__ rows,
                    const int* __restrict__ cols,
                    const float* __restrict__ vals,
                    const float* __restrict__ embeds,
                    float* __restrict__ out,
                    int nEdges) {
  __shared__ int   s_rows[TILE];
  __shared__ int   s_cols[TILE];
  __shared__ float s_vals[TILE];

  const int base = blockIdx.x * TILE;
  const int t    = threadIdx.x;

  // ---- Stage this block's edge tile into LDS with async global->LDS B128 ----
  {
    const int g = base + 4 * t;              // 4 elements per thread per array
    if (g + 4 <= nEdges) {                   // E % 4 == 0, so chunks never straddle
      __builtin_amdgcn_global_load_async_to_lds_b128(
          (v4i*)(rows + g), (lv4i*)(&s_rows[4 * t]), 0, 0);
      __builtin_amdgcn_global_load_async_to_lds_b128(
          (v4i*)(cols + g), (lv4i*)(&s_cols[4 * t]), 0, 0);
      __builtin_amdgcn_global_load_async_to_lds_b128(
          (v4i*)(vals + g), (lv4i*)(&s_vals[4 * t]), 0, 0);
    }
  }
  asm volatile("s_wait_asynccnt 0" ::: "memory");
  __syncthreads();

  const int nE   = min(TILE, nEdges - base);
  const int wave = t >> 5;         // 8 waves per block
  const int lane = t & 31;
  const int sub  = lane >> 4;      // which of 2 edges this half-wave handles
  const int q4   = (lane & 15) << 2;  // float offset of this lane's float4 chunk

  const int eBase = wave * (TILE / 8) + sub;   // each wave owns 128 edges
  const float* __restrict__ embQ = embeds + q4;
  float* __restrict__ outQ = out + q4;

  if (nE == TILE) {
    // Fast path: full tile, no per-iteration bounds check in the hot loop.
    #pragma unroll 8
    for (int i = 0; i < TILE / 16; ++i) {      // 64 iters x 2 edges per wave
      const int e = eBase + i * 2;
      const int   r = s_rows[e];
      const int   c = s_cols[e];
      const float v = s_vals[e];
      // Coalesced 256B gather: 16 lanes x 16B = one embedding row.
      const float4 m = *(const float4*)(embQ + (size_t)c * 64);
      float* dst = outQ + (size_t)r * 64;
      // Fire-and-forget fp32 atomics (no return -> STOREcnt path).
      __hip_atomic_fetch_add(dst + 0, v * m.x, __ATOMIC_RELAXED, __HIP_MEMORY_SCOPE_AGENT);
      __hip_atomic_fetch_add(dst + 1, v * m.y, __ATOMIC_RELAXED, __HIP_MEMORY_SCOPE_AGENT);
      __hip_atomic_fetch_add(dst + 2, v * m.z, __ATOMIC_RELAXED, __HIP_MEMORY_SCOPE_AGENT);
      __hip_atomic_fetch_add(dst + 3, v * m.w, __ATOMIC_RELAXED, __HIP_MEMORY_SCOPE_AGENT);
    }
  } else {
    // Tail path: partial tile (last block only).
    #pragma unroll 4
    for (int i = 0; i < TILE / 16; ++i) {
      const int e = eBase + i * 2;
      if (e < nE) {
        const int   r = s_rows[e];
        const int   c = s_cols[e];
        const float v = s_vals[e];
        const float4 m = *(const float4*)(embQ + (size_t)c * 64);
        float* dst = outQ + (size_t)r * 64;
        __hip_atomic_fetch_add(dst + 0, v * m.x, __ATOMIC_RELAXED, __HIP_MEMORY_SCOPE_AGENT);
        __hip_atomic_fetch_add(dst + 1, v * m.y, __ATOMIC_RELAXED, __HIP_MEMORY_SCOPE_AGENT);
        __hip_atomic_fetch_add(dst + 2, v * m.z, __ATOMIC_RELAXED, __HIP_MEMORY_SCOPE_AGENT);
        __hip_atomic_fetch_add(dst + 3, v * m.w, __ATOMIC_RELAXED, __HIP_MEMORY_SCOPE_AGENT);
      }
    }
  }
}

extern "C" void kernel_launch(void* const* d_in, const int* in_sizes, int n_in,
                              void* d_out, int out_size, void* d_ws, size_t ws_size,
                              hipStream_t stream) {
  const int*   rows   = (const int*)d_in[0];
  const int*   cols   = (const int*)d_in[1];
  const float* vals   = (const float*)d_in[2];
  const float* embeds = (const float*)d_in[3];
  // d_in[4] = n_nodes scalar; implied by out_size = n_nodes * 64
  const int nEdges = in_sizes[0];
  float* out = (float*)d_out;

  // Must zero the accumulator every launch (kernel accumulates via atomics).
  const int n4 = out_size / 4;
  gcn_zero_kernel<<<(n4 + 255) / 256, 256, 0, stream>>>((float4*)out, n4);

  const int nTiles = (nEdges + TILE - 1) / TILE;
  gcn_agg_kernel<<<nTiles, THREADS, 0, stream>>>(rows, cols, vals, embeds, out, nEdges);
}